// IPU_Criterion_85804856639872
// MI455X (gfx1250) — compile-verified
//
#include <hip/hip_runtime.h>

typedef __attribute__((ext_vector_type(2))) float v2f;
typedef __attribute__((ext_vector_type(8))) float v8f;

union V8 { v8f v; float f[8]; };
union V2 { v2f v; float f[2]; };

__device__ inline v8f wmma4(v2f a, v2f b, v8f c) {
  // D(16x16,f32) = A(16x4,f32) * B(4x16,f32) + C  -> v_wmma_f32_16x16x4_f32
  return __builtin_amdgcn_wmma_f32_16x16x4_f32(false, a, false, b, (short)0, c,
                                               false, false);
}

// ---------------------------------------------------------------------------
// Pack points for WMMA distance tiles:
//   rowp[i] = (x, y, z, 1)          (A-matrix rows)
//   colp[i] = (-2x, -2y, -2z, |p|^2)  (B-matrix cols)
//   sq[i]   = |p|^2                  (C accumulator init for the row set)
// ---------------------------------------------------------------------------
__global__ void pack_kernel(const float* __restrict__ src, int stride, int n,
                            float4* __restrict__ rowp, float4* __restrict__ colp,
                            float* __restrict__ sq) {
  int i = blockIdx.x * 256 + threadIdx.x;
  if (i >= n) return;
  const float* s = src + (size_t)i * stride;
  float x = s[0], y = s[1], z = s[2];
  float q = x * x + y * y + z * z;
  rowp[i] = make_float4(x, y, z, 1.0f);
  colp[i] = make_float4(-2.f * x, -2.f * y, -2.f * z, q);
  sq[i] = q;
}

// ---------------------------------------------------------------------------
// Per-wave 16-row block, scan all column tiles with one WMMA each.
// C layout (f32 16x16): VGPR v, lane l -> M = v + 8*(l>=16), N = l%16.
// ---------------------------------------------------------------------------
__global__ __launch_bounds__(32)
void rowmin_kernel(const float4* __restrict__ rowP, const float* __restrict__ sqP,
                   const float4* __restrict__ colQ, int np, int nq,
                   float* __restrict__ outmin) {
  const int b = blockIdx.y;
  const int rowbase = blockIdx.x * 16;
  const int lane = threadIdx.x, half = lane >> 4, l16 = lane & 15;
  const float4* Pr = rowP + (size_t)b * np;
  const float*  Ps = sqP  + (size_t)b * np;
  const float4* Qc = colQ + (size_t)b * nq;

  float2 ap = ((const float2*)(Pr + rowbase + l16))[half];
  V2 a; a.f[0] = ap.x; a.f[1] = ap.y;
  const float4* ps4 = (const float4*)(Ps + rowbase + 8 * half);
  float4 s0 = ps4[0], s1 = ps4[1];
  V8 cinit;
  cinit.f[0] = s0.x; cinit.f[1] = s0.y; cinit.f[2] = s0.z; cinit.f[3] = s0.w;
  cinit.f[4] = s1.x; cinit.f[5] = s1.y; cinit.f[6] = s1.z; cinit.f[7] = s1.w;

  float rm[8];
#pragma unroll
  for (int v = 0; v < 8; v++) rm[v] = 3.4e38f;

  for (int j0 = 0; j0 < nq; j0 += 16) {
    float2 bp = ((const float2*)(Qc + j0 + l16))[half];
    V2 bb; bb.f[0] = bp.x; bb.f[1] = bp.y;
    V8 d; d.v = wmma4(a.v, bb.v, cinit.v);
#pragma unroll
    for (int v = 0; v < 8; v++) rm[v] = fminf(rm[v], d.f[v]);
  }
#pragma unroll
  for (int off = 8; off >= 1; off >>= 1) {
#pragma unroll
    for (int v = 0; v < 8; v++) rm[v] = fminf(rm[v], __shfl_xor(rm[v], off, 16));
  }
  if (l16 == 0) {
#pragma unroll
    for (int v = 0; v < 8; v++)
      outmin[(size_t)b * np + rowbase + 8 * half + v] = rm[v];
  }
}

__global__ __launch_bounds__(32)
void rowargmin_kernel(const float4* __restrict__ rowP, const float* __restrict__ sqP,
                      const float4* __restrict__ colQ, int np, int nq,
                      int* __restrict__ outidx) {
  const int b = blockIdx.y;
  const int rowbase = blockIdx.x * 16;
  const int lane = threadIdx.x, half = lane >> 4, l16 = lane & 15;
  const float4* Pr = rowP + (size_t)b * np;
  const float*  Ps = sqP  + (size_t)b * np;
  const float4* Qc = colQ + (size_t)b * nq;

  float2 ap = ((const float2*)(Pr + rowbase + l16))[half];
  V2 a; a.f[0] = ap.x; a.f[1] = ap.y;
  const float4* ps4 = (const float4*)(Ps + rowbase + 8 * half);
  float4 s0 = ps4[0], s1 = ps4[1];
  V8 cinit;
  cinit.f[0] = s0.x; cinit.f[1] = s0.y; cinit.f[2] = s0.z; cinit.f[3] = s0.w;
  cinit.f[4] = s1.x; cinit.f[5] = s1.y; cinit.f[6] = s1.z; cinit.f[7] = s1.w;

  float rm[8]; int ri[8];
#pragma unroll
  for (int v = 0; v < 8; v++) { rm[v] = 3.4e38f; ri[v] = 0; }

  for (int j0 = 0; j0 < nq; j0 += 16) {
    float2 bp = ((const float2*)(Qc + j0 + l16))[half];
    V2 bb; bb.f[0] = bp.x; bb.f[1] = bp.y;
    V8 d; d.v = wmma4(a.v, bb.v, cinit.v);
    int j = j0 + l16;
#pragma unroll
    for (int v = 0; v < 8; v++)
      if (d.f[v] < rm[v]) { rm[v] = d.f[v]; ri[v] = j; }
  }
#pragma unroll
  for (int off = 8; off >= 1; off >>= 1) {
#pragma unroll
    for (int v = 0; v < 8; v++) {
      float ov = __shfl_xor(rm[v], off, 16);
      int   oi = __shfl_xor(ri[v], off, 16);
      if (ov < rm[v]) { rm[v] = ov; ri[v] = oi; }
    }
  }
  if (l16 == 0) {
#pragma unroll
    for (int v = 0; v < 8; v++)
      outidx[(size_t)b * np + rowbase + 8 * half + v] = ri[v];
  }
}

// ---------------------------------------------------------------------------
// KNN (K=6 smallest sqdist, self included) for pts vs pts.
// Single-wave workgroup: tile bounced through LDS, per-lane sorted 6-list
// over one row-half, pairwise merge at the end.
// ---------------------------------------------------------------------------
__global__ __launch_bounds__(32)
void knn_kernel(const float4* __restrict__ rowP, const float* __restrict__ sqP,
                const float4* __restrict__ colP, int n, int* __restrict__ knnOut) {
  __shared__ float tile[16][17];
  __shared__ float mv[32][6];
  __shared__ int   mi[32][6];
  const int b = blockIdx.y;
  const int rowbase = blockIdx.x * 16;
  const int lane = threadIdx.x, half = lane >> 4, l16 = lane & 15;
  const float4* Pr = rowP + (size_t)b * n;
  const float*  Ps = sqP  + (size_t)b * n;
  const float4* Qc = colP + (size_t)b * n;

  float2 ap = ((const float2*)(Pr + rowbase + l16))[half];
  V2 a; a.f[0] = ap.x; a.f[1] = ap.y;
  const float4* ps4 = (const float4*)(Ps + rowbase + 8 * half);
  float4 s0 = ps4[0], s1 = ps4[1];
  V8 cinit;
  cinit.f[0] = s0.x; cinit.f[1] = s0.y; cinit.f[2] = s0.z; cinit.f[3] = s0.w;
  cinit.f[4] = s1.x; cinit.f[5] = s1.y; cinit.f[6] = s1.z; cinit.f[7] = s1.w;

  float kv[6]; int ki[6];
#pragma unroll
  for (int k = 0; k < 6; k++) { kv[k] = 3.4e38f; ki[k] = 0; }

  for (int j0 = 0; j0 < n; j0 += 16) {
    float2 bp = ((const float2*)(Qc + j0 + l16))[half];
    V2 bb; bb.f[0] = bp.x; bb.f[1] = bp.y;
    V8 d; d.v = wmma4(a.v, bb.v, cinit.v);
#pragma unroll
    for (int v = 0; v < 8; v++) tile[8 * half + v][l16] = d.f[v];
    __syncthreads();
    const int r = l16, c0 = half * 8;
#pragma unroll
    for (int cc = 0; cc < 8; cc++) {
      float val = tile[r][c0 + cc];
      if (val < kv[5]) {
        int idx = j0 + c0 + cc;
        int p = 5;
        while (p > 0 && kv[p - 1] > val) {
          kv[p] = kv[p - 1]; ki[p] = ki[p - 1]; --p;
        }
        kv[p] = val; ki[p] = idx;
      }
    }
    __syncthreads();
  }
#pragma unroll
  for (int k = 0; k < 6; k++) { mv[lane][k] = kv[k]; mi[lane][k] = ki[k]; }
  __syncthreads();
  if (lane < 16) {
    int ia = 0, ib = 0;
    int* out = knnOut + ((size_t)b * n + rowbase + lane) * 6;
    for (int k = 0; k < 6; k++) {
      float va = mv[lane][ia], vb = mv[lane + 16][ib];
      if (va <= vb) { out[k] = mi[lane][ia]; ia++; }
      else          { out[k] = mi[lane + 16][ib]; ib++; }
    }
  }
}

// ---------------------------------------------------------------------------
// Point-wise epilogue
// ---------------------------------------------------------------------------
__device__ inline float3 bnorm3(float x, float y, float z) {
  float ss = x * x + y * y + z * z;
  float inv = 1.0f / (sqrtf(ss + 1e-8f) + 1e-10f);
  return make_float3(x * inv, y * inv, z * inv);
}
__device__ inline float cossim3(float3 a, float3 b) {
  float na = fmaxf(sqrtf(a.x * a.x + a.y * a.y + a.z * a.z), 1e-8f);
  float nb = fmaxf(sqrtf(b.x * b.x + b.y * b.y + b.z * b.z), 1e-8f);
  return (a.x * b.x + a.y * b.y + a.z * b.z) / (na * nb);
}
__device__ inline void waveRedAdd(float v, float* slot) {
#pragma unroll
  for (int off = 16; off >= 1; off >>= 1) v += __shfl_down(v, off, 32);
  if ((threadIdx.x & 31) == 0) atomicAdd(slot, v);
}

__global__ void point_losses(const float* __restrict__ ori_pre,
                             const float* __restrict__ nor_pre,
                             const float* __restrict__ gt,
                             const int* __restrict__ nnidx,
                             float* __restrict__ norg, float* __restrict__ oripro,
                             int N, int NG, float* __restrict__ acc) {
  int gid = blockIdx.x * 256 + threadIdx.x;  // over B*N (exact grid)
  int b = gid / N;
  int nn = nnidx[gid];
  const float* g = gt + ((size_t)b * NG + nn) * 6;
  float3 ng = make_float3(g[3], g[4], g[5]);
  const float* op = ori_pre + (size_t)gid * 3;
  float3 on = bnorm3(op[0], op[1], op[2]);
  float dn = on.x * ng.x + on.y * ng.y + on.z * ng.z;
  float3 pr = bnorm3(on.x - ng.x * dn, on.y - ng.y * dn, on.z - ng.z * dn);
  norg[(size_t)gid * 3 + 0] = ng.x; norg[(size_t)gid * 3 + 1] = ng.y; norg[(size_t)gid * 3 + 2] = ng.z;
  oripro[(size_t)gid * 3 + 0] = pr.x; oripro[(size_t)gid * 3 + 1] = pr.y; oripro[(size_t)gid * 3 + 2] = pr.z;
  const float* nq = nor_pre + (size_t)gid * 3;
  float3 nb = bnorm3(nq[0], nq[1], nq[2]);
  float ln = 1.0f - fabsf(cossim3(ng, nb));
  float lno = fabsf(dn);
  waveRedAdd(ln, acc + 1);
  waveRedAdd(lno, acc + 2);
}

__global__ void smooth_kernel(const float* __restrict__ oripro,
                              const float* __restrict__ norg,
                              const int* __restrict__ knnIdx, int N,
                              float* __restrict__ acc) {
  int gid = blockIdx.x * 256 + threadIdx.x;  // over B*N
  int b = gid / N;
  float3 bo = make_float3(oripro[(size_t)gid * 3], oripro[(size_t)gid * 3 + 1], oripro[(size_t)gid * 3 + 2]);
  float3 bn = make_float3(norg[(size_t)gid * 3], norg[(size_t)gid * 3 + 1], norg[(size_t)gid * 3 + 2]);
  float3 br = make_float3(bo.y * bn.z - bo.z * bn.y,
                          bo.z * bn.x - bo.x * bn.z,
                          bo.x * bn.y - bo.y * bn.x);
  float s = 0.f;
#pragma unroll
  for (int k = 0; k < 6; k++) {
    int j = knnIdx[(size_t)gid * 6 + k];
    size_t o = ((size_t)b * N + j) * 3;
    float3 go = make_float3(oripro[o], oripro[o + 1], oripro[o + 2]);
    float3 gn = make_float3(norg[o], norg[o + 1], norg[o + 2]);
    float nd = gn.x * bn.x + gn.y * bn.y + gn.z * bn.z;
    float w = expf(-nd / 0.3f) * 10.f + 1.f;
    w = (w < 4.f) ? 1.f : 5.f;
    float c0 = cossim3(go, bo), c1 = cossim3(go, br);
    float c = fminf(1.f - fabsf(c0), 1.f - fabsf(c1));
    s += w * c;
  }
  waveRedAdd(s, acc + 0);
}

__global__ void sum_kernel(const float* __restrict__ x, int n, float* __restrict__ slot) {
  float s = 0.f;
  for (int i = blockIdx.x * 256 + threadIdx.x; i < n; i += gridDim.x * 256) s += x[i];
  waveRedAdd(s, slot);
}

__global__ void zero_acc(float* acc) { if (threadIdx.x < 8) acc[threadIdx.x] = 0.f; }

__global__ void final_kernel(const float* __restrict__ acc, float* __restrict__ out,
                             int N, int NG, int NU, int B) {
  float smooth   = acc[0] / ((float)B * N * 6);
  float lnor     = acc[1] / ((float)B * N);
  float lnorori  = acc[2] / ((float)B * N);
  float charm    = acc[3] / ((float)B * NU) + acc[4] / ((float)B * NG);
  float charmoff = acc[5] / ((float)B * NU) + acc[6] / ((float)B * NG);
  float cd = charm + 0.4f * charmoff;
  float loss = smooth + lnor + 0.1f * lnorori + 200.f * cd;
  out[0] = loss; out[1] = smooth; out[2] = lnor; out[3] = charmoff;
  out[4] = lnorori; out[5] = charm; out[6] = charm;
}

// ---------------------------------------------------------------------------
extern "C" void kernel_launch(void* const* d_in, const int* in_sizes, int n_in,
                              void* d_out, int out_size, void* d_ws, size_t ws_size,
                              hipStream_t stream) {
  const float* ori_pre = (const float*)d_in[0];
  const float* nor_pre = (const float*)d_in[1];
  const float* xyz_up  = (const float*)d_in[2];
  const float* xyz_off = (const float*)d_in[3];
  const float* pts     = (const float*)d_in[4];
  const float* gt      = (const float*)d_in[5];

  const int B  = 4;
  const int N  = in_sizes[0] / (B * 3);
  const int NU = in_sizes[2] / (B * 3);
  const int NG = in_sizes[5] / (B * 6);

  size_t off = 0;
  auto alloc = [&](size_t bytes) -> void* {
    off = (off + 255) & ~(size_t)255;
    void* p = (char*)d_ws + off;
    off += bytes;
    return p;
  };
  float4* rp_pts = (float4*)alloc((size_t)B * N * 16);
  float4* cp_pts = (float4*)alloc((size_t)B * N * 16);
  float*  sq_pts = (float*) alloc((size_t)B * N * 4);
  float4* rp_gt  = (float4*)alloc((size_t)B * NG * 16);
  float4* cp_gt  = (float4*)alloc((size_t)B * NG * 16);
  float*  sq_gt  = (float*) alloc((size_t)B * NG * 4);
  float4* rp_up  = (float4*)alloc((size_t)B * NU * 16);
  float4* cp_up  = (float4*)alloc((size_t)B * NU * 16);
  float*  sq_up  = (float*) alloc((size_t)B * NU * 4);
  float4* rp_of  = (float4*)alloc((size_t)B * NU * 16);
  float4* cp_of  = (float4*)alloc((size_t)B * NU * 16);
  float*  sq_of  = (float*) alloc((size_t)B * NU * 4);
  int*    nnidx  = (int*)   alloc((size_t)B * N * 4);
  int*    knnIdx = (int*)   alloc((size_t)B * N * 6 * 4);
  float*  norg   = (float*) alloc((size_t)B * N * 12);
  float*  oripro = (float*) alloc((size_t)B * N * 12);
  float*  rmin_up = (float*)alloc((size_t)B * NU * 4);
  float*  cmin_up = (float*)alloc((size_t)B * NG * 4);
  float*  rmin_of = (float*)alloc((size_t)B * NU * 4);
  float*  cmin_of = (float*)alloc((size_t)B * NG * 4);
  float*  acc     = (float*)alloc(32);

  zero_acc<<<1, 32, 0, stream>>>(acc);

  pack_kernel<<<(B * N  + 255) / 256, 256, 0, stream>>>(pts,     3, B * N,  rp_pts, cp_pts, sq_pts);
  pack_kernel<<<(B * NG + 255) / 256, 256, 0, stream>>>(gt,      6, B * NG, rp_gt,  cp_gt,  sq_gt);
  pack_kernel<<<(B * NU + 255) / 256, 256, 0, stream>>>(xyz_up,  3, B * NU, rp_up,  cp_up,  sq_up);
  pack_kernel<<<(B * NU + 255) / 256, 256, 0, stream>>>(xyz_off, 3, B * NU, rp_of,  cp_of,  sq_of);

  rowargmin_kernel<<<dim3(N / 16, B), 32, 0, stream>>>(rp_pts, sq_pts, cp_gt, N, NG, nnidx);
  knn_kernel<<<dim3(N / 16, B), 32, 0, stream>>>(rp_pts, sq_pts, cp_pts, N, knnIdx);

  rowmin_kernel<<<dim3(NU / 16, B), 32, 0, stream>>>(rp_up, sq_up, cp_gt, NU, NG, rmin_up);
  rowmin_kernel<<<dim3(NG / 16, B), 32, 0, stream>>>(rp_gt, sq_gt, cp_up, NG, NU, cmin_up);
  rowmin_kernel<<<dim3(NU / 16, B), 32, 0, stream>>>(rp_of, sq_of, cp_gt, NU, NG, rmin_of);
  rowmin_kernel<<<dim3(NG / 16, B), 32, 0, stream>>>(rp_gt, sq_gt, cp_of, NG, NU, cmin_of);

  point_losses<<<(B * N) / 256, 256, 0, stream>>>(ori_pre, nor_pre, gt, nnidx,
                                                  norg, oripro, N, NG, acc);
  smooth_kernel<<<(B * N) / 256, 256, 0, stream>>>(oripro, norg, knnIdx, N, acc);

  sum_kernel<<<64, 256, 0, stream>>>(rmin_up, B * NU, acc + 3);
  sum_kernel<<<64, 256, 0, stream>>>(cmin_up, B * NG, acc + 4);
  sum_kernel<<<64, 256, 0, stream>>>(rmin_of, B * NU, acc + 5);
  sum_kernel<<<64, 256, 0, stream>>>(cmin_of, B * NG, acc + 6);

  final_kernel<<<1, 1, 0, stream>>>(acc, (float*)d_out, N, NG, NU, B);
}